// YoloModel_13657996001588
// MI455X (gfx1250) — compile-verified
//
#include <hip/hip_runtime.h>
#include <stdint.h>

#define N_ANCH     100800
#define NC         80
#define NPRED      85
#define CONF_T     0.4f
#define IOU_T      0.45f
#define MAX_DET    1000
#define MAX_NMS    4096
#define MAX_WH     7680.0f
#define MASK_WORDS 128          // 4096 bits / 32
#define SCORE_BLK  128          // anchors staged per block (43.5 KB LDS)

typedef float v2f __attribute__((ext_vector_type(2)));
typedef float v8f __attribute__((ext_vector_type(8)));

struct Ctrl {
  unsigned int        hist[256];
  unsigned long long  prefix;
  unsigned int        k;
  unsigned int        counter;
};

__device__ __forceinline__ unsigned int mapf(float f) {
  unsigned int u = __float_as_uint(f);
  return (u & 0x80000000u) ? ~u : (u | 0x80000000u);
}

// ---------------- init scratch ----------------
__global__ void init_kernel(unsigned int* mask, Ctrl* ctrl, int maskWords) {
  int i = blockIdx.x * blockDim.x + threadIdx.x;
  if (i < maskWords) mask[i] = 0u;        // lower-triangle halfwords must be 0
  if (i < 256) ctrl->hist[i] = 0u;
  if (i == 0) { ctrl->prefix = 0ull; ctrl->k = MAX_NMS; ctrl->counter = 0u; }
}

// ---------------- stage 1: LDS-staged scores, boxes, sort keys ----------------
__global__ void __launch_bounds__(SCORE_BLK) score_kernel(const float* __restrict__ pred,
                                                          unsigned long long* __restrict__ keys,
                                                          float* __restrict__ aux) {
  __shared__ float tile[SCORE_BLK * NPRED];   // 128*85*4 = 43520 B
  int base = blockIdx.x * SCORE_BLK;
  int count = N_ANCH - base; if (count > SCORE_BLK) count = SCORE_BLK;
  int total = count * NPRED;
  const float* src = pred + (size_t)base * NPRED;
  for (int t = threadIdx.x; t < total; t += SCORE_BLK) tile[t] = src[t];  // coalesced
  __syncthreads();

  int li = threadIdx.x;
  if (li >= count) return;
  const float* p = tile + li * NPRED;         // odd stride -> bank-conflict-free
  float x = p[0], y = p[1], w = p[2], h = p[3], obj = p[4];
  float best = -1.0f; int bj = 0;
  for (int j = 0; j < NC; ++j) {
    float v = p[5 + j] * obj;                 // cls_conf = cls * obj (ref order)
    if (v > best) { best = v; bj = j; }       // first-max == jnp.argmax
  }
  float score = (best > CONF_T) ? best : -__builtin_inff();
  int i = base + li;
  keys[i] = ((unsigned long long)(~mapf(score)) << 32) | (unsigned int)i;
  float* a = aux + (size_t)i * 8;
  a[0] = x - 0.5f * w; a[1] = y - 0.5f * h;
  a[2] = x + 0.5f * w; a[3] = y + 0.5f * h;
  a[4] = score; a[5] = (float)bj; a[6] = 0.0f; a[7] = 0.0f;
}

// ---------------- radix-select: LDS byte histogram, one merge per block ----------------
__global__ void __launch_bounds__(256) hist_kernel(const unsigned long long* __restrict__ keys,
                                                   Ctrl* ctrl, int r) {
  __shared__ unsigned int h[256];
  h[threadIdx.x] = 0u;
  __syncthreads();
  unsigned long long pfx = ctrl->prefix;
  int stride = gridDim.x * blockDim.x;
  for (int i = blockIdx.x * blockDim.x + threadIdx.x; i < N_ANCH; i += stride) {
    unsigned long long key = keys[i];
    bool match = (r == 7) || ((key >> ((r + 1) * 8)) == (pfx >> ((r + 1) * 8)));
    if (match)
      atomicAdd(&h[(unsigned int)(key >> (r * 8)) & 0xFFu], 1u);
  }
  __syncthreads();
  unsigned int c = h[threadIdx.x];
  if (c) atomicAdd(&ctrl->hist[threadIdx.x], c);
}

// ---------------- radix-select: pick byte of k-th smallest ----------------
__global__ void select_kernel(Ctrl* ctrl, int r) {
  __shared__ unsigned int h[256];
  h[threadIdx.x] = ctrl->hist[threadIdx.x];
  __syncthreads();
  if (threadIdx.x == 0) {
    unsigned int k = ctrl->k;
    unsigned long long cum = 0;
    int bsel = 255;
    for (int b = 0; b < 256; ++b) {
      unsigned int c = h[b];
      if (cum + c >= (unsigned long long)k) { bsel = b; break; }
      cum += c;
    }
    ctrl->prefix |= ((unsigned long long)bsel) << (r * 8);
    ctrl->k = k - (unsigned int)cum;
  }
  __syncthreads();
  ctrl->hist[threadIdx.x] = 0u;   // ready for next round
}

// ---------------- compact the exact top-4096 keys ----------------
__global__ void compact_kernel(const unsigned long long* __restrict__ keys,
                               Ctrl* ctrl, unsigned long long* __restrict__ sel) {
  unsigned long long kth = ctrl->prefix;   // exact 4096-th smallest key
  int stride = gridDim.x * blockDim.x;
  for (int i = blockIdx.x * blockDim.x + threadIdx.x; i < N_ANCH; i += stride) {
    unsigned long long key = keys[i];
    if (key <= kth) {
      unsigned int pos = atomicAdd(&ctrl->counter, 1u);
      if (pos < MAX_NMS) sel[pos] = key;
    }
  }
}

// ---------------- single-WG bitonic sort in LDS (32 KB of 320 KB) ----------------
__global__ void __launch_bounds__(1024) sort_kernel(unsigned long long* sel) {
  __shared__ unsigned long long s[MAX_NMS];
  for (int i = threadIdx.x; i < MAX_NMS; i += 1024) s[i] = sel[i];
  __syncthreads();
  for (int k = 2; k <= MAX_NMS; k <<= 1) {
    for (int j = k >> 1; j > 0; j >>= 1) {
      for (int t = threadIdx.x; t < MAX_NMS; t += 1024) {
        int ixj = t ^ j;
        if (ixj > t) {
          bool asc = ((t & k) == 0);
          unsigned long long a = s[t], b = s[ixj];
          if (asc ? (a > b) : (a < b)) { s[t] = b; s[ixj] = a; }
        }
      }
      __syncthreads();
    }
  }
  for (int i = threadIdx.x; i < MAX_NMS; i += 1024) sel[i] = s[i];
}

// ---------------- gather selected boxes into SoA ----------------
__global__ void gather_kernel(const unsigned long long* __restrict__ sel,
                              const float* __restrict__ aux,
                              float* __restrict__ soa) {
  int k = blockIdx.x * blockDim.x + threadIdx.x;
  if (k >= MAX_NMS) return;
  unsigned long long key = sel[k];
  unsigned int idx = (unsigned int)(key & 0xFFFFFFFFu);
  const float* a = aux + (size_t)idx * 8;
  float x1 = a[0], y1 = a[1], x2 = a[2], y2 = a[3], sc = a[4], cf = a[5];
  float off = cf * MAX_WH;                 // class-offset trick
  soa[0 * MAX_NMS + k] = x1 + off;
  soa[1 * MAX_NMS + k] = y1 + off;
  soa[2 * MAX_NMS + k] = x2 + off;
  soa[3 * MAX_NMS + k] = y2 + off;
  soa[4 * MAX_NMS + k] = (x2 - x1) * (y2 - y1);   // area (offset-invariant)
  soa[5 * MAX_NMS + k] = sc;
  soa[6 * MAX_NMS + k] = cf;
  soa[7 * MAX_NMS + k] = x1;
  soa[8 * MAX_NMS + k] = y1;
  soa[9 * MAX_NMS + k] = x2;
  soa[10 * MAX_NMS + k] = y2;
}

// ---------------- IoU tiles: WMMA outer-sum + ballot bit-pack, no atomics ----------------
// 4 waves/block, each wave owns one (it, jt) 16x16 tile. Wave-uniform early-out
// keeps EXEC all-ones for the WMMA in active waves.
__global__ void __launch_bounds__(128) iou_kernel(const float* __restrict__ soa,
                                                  unsigned short* __restrict__ mask16) {
  int it = blockIdx.y;
  int jt = blockIdx.x * 4 + (threadIdx.x >> 5);
  if (jt < it) return;                     // strictly-lower tiles have no j>i bits
  int lane = threadIdx.x & 31;
  int n = lane & 15;
  const float* obx1 = soa + 0 * MAX_NMS;
  const float* oby1 = soa + 1 * MAX_NMS;
  const float* obx2 = soa + 2 * MAX_NMS;
  const float* oby2 = soa + 3 * MAX_NMS;
  const float* areap = soa + 4 * MAX_NMS;

  int jg = jt * 16 + n;
  float jx1 = obx1[jg], jy1 = oby1[jg], jx2 = obx2[jg], jy2 = oby2[jg];
  float aj = areap[jg];
  bool lo = lane < 16;

  // A (16x4): col0 = area_i, col1 = 1, cols2-3 = 0
  v2f a; a.x = lo ? areap[it * 16 + n] : 0.0f; a.y = lo ? 1.0f : 0.0f;
  // B (4x16): row0 = 1, row1 = area_j, rows2-3 = 0
  v2f b; b.x = lo ? 1.0f : 0.0f; b.y = lo ? aj : 0.0f;
  v8f c = {0.0f, 0.0f, 0.0f, 0.0f, 0.0f, 0.0f, 0.0f, 0.0f};
  // D[m][n] = area_i[m] + area_j[n]
  v8f d = __builtin_amdgcn_wmma_f32_16x16x4_f32(false, a, false, b,
                                                (short)0, c, false, false);
#pragma unroll
  for (int v = 0; v < 8; ++v) {
    int m = it * 16 + (lo ? v : v + 8);    // 16x16 f32 C/D VGPR layout
    float ix1 = obx1[m], iy1 = oby1[m], ix2 = obx2[m], iy2 = oby2[m];
    float ww = fmaxf(fminf(ix2, jx2) - fmaxf(ix1, jx1), 0.0f);
    float hh = fmaxf(fminf(iy2, jy2) - fmaxf(iy1, jy1), 0.0f);
    float inter = ww * hh;
    float iou = inter / (d[v] - inter + 1e-9f);
    bool sup = (iou > IOU_T) && (jg > m);
    unsigned int bal = (unsigned int)__ballot(sup);   // wave32 mask, all lanes
    // each (row, jt) halfword has exactly one writer -> plain stores
    if (lane == 0)
      mask16[(size_t)(it * 16 + v) * (MASK_WORDS * 2) + jt] =
          (unsigned short)(bal & 0xFFFFu);             // row v    (lanes 0-15)
    if (lane == 16)
      mask16[(size_t)(it * 16 + v + 8) * (MASK_WORDS * 2) + jt] =
          (unsigned short)(bal >> 16);                 // row v+8  (lanes 16-31)
  }
}

// ---------------- sequential greedy suppression + emit top-1000 ----------------
__global__ void __launch_bounds__(128) nms_out_kernel(const unsigned int* __restrict__ mask,
                                                      const float* __restrict__ soa,
                                                      float* __restrict__ out) {
  __shared__ unsigned int keep[MASK_WORDS];
  __shared__ unsigned int pfx[MASK_WORDS];
  int tid = threadIdx.x;
  const float* scorep = soa + 5 * MAX_NMS;

  unsigned int w = 0u;                     // keep init = valid = score > thres
  for (int bbit = 0; bbit < 32; ++bbit) {
    int j = tid * 32 + bbit;
    if (scorep[j] > CONF_T) w |= (1u << bbit);
  }
  keep[tid] = w;
  __syncthreads();

  for (int i = 0; i < MAX_NMS; ++i) {
    unsigned int kw = keep[i >> 5];
    if ((kw >> (i & 31)) & 1u)
      keep[tid] &= ~mask[(size_t)i * MASK_WORDS + tid];
    if (i + 1 < MAX_NMS)                   // hide next row's L2 latency
      __builtin_prefetch(&mask[(size_t)(i + 1) * MASK_WORDS + tid], 0, 1);
    __syncthreads();
  }

  for (int x = tid; x < MAX_DET * 6; x += 128) out[x] = 0.0f;
  if (tid == 0) {
    unsigned int cacc = 0;
    for (int q = 0; q < MASK_WORDS; ++q) { pfx[q] = cacc; cacc += __popc(keep[q]); }
  }
  __syncthreads();

  const float* clsp = soa + 6 * MAX_NMS;
  const float* rbx1 = soa + 7 * MAX_NMS;
  const float* rby1 = soa + 8 * MAX_NMS;
  const float* rbx2 = soa + 9 * MAX_NMS;
  const float* rby2 = soa + 10 * MAX_NMS;
  for (int j = tid; j < MAX_NMS; j += 128) {
    unsigned int kw = keep[j >> 5];
    if ((kw >> (j & 31)) & 1u) {
      unsigned int rank = pfx[j >> 5] + __popc(kw & ((1u << (j & 31)) - 1u));
      if (rank < MAX_DET) {
        float* o = out + (size_t)rank * 6;
        o[0] = rbx1[j]; o[1] = rby1[j]; o[2] = rbx2[j]; o[3] = rby2[j];
        o[4] = scorep[j]; o[5] = clsp[j];
      }
    }
  }
}

extern "C" void kernel_launch(void* const* d_in, const int* in_sizes, int n_in,
                              void* d_out, int out_size, void* d_ws, size_t ws_size,
                              hipStream_t stream) {
  (void)in_sizes; (void)n_in; (void)out_size; (void)ws_size;
  const float* pred = (const float*)d_in[0];
  float* out = (float*)d_out;
  char* ws = (char*)d_ws;

  size_t o_keys = 0;
  size_t o_aux  = o_keys + (size_t)N_ANCH * 8;              // 64-bit keys
  size_t o_sel  = o_aux  + (size_t)N_ANCH * 8 * 4;          // 8 floats / anchor
  size_t o_soa  = o_sel  + (size_t)MAX_NMS * 8;
  size_t o_mask = o_soa  + (size_t)11 * MAX_NMS * 4;
  size_t o_ctrl = o_mask + (size_t)MAX_NMS * MASK_WORDS * 4;

  unsigned long long* keys = (unsigned long long*)(ws + o_keys);
  float*              aux  = (float*)(ws + o_aux);
  unsigned long long* sel  = (unsigned long long*)(ws + o_sel);
  float*              soa  = (float*)(ws + o_soa);
  unsigned int*       mask = (unsigned int*)(ws + o_mask);
  unsigned short*     mask16 = (unsigned short*)(ws + o_mask);
  Ctrl*               ctrl = (Ctrl*)(ws + o_ctrl);

  int maskWords = MAX_NMS * MASK_WORDS;
  init_kernel<<<(maskWords + 255) / 256, 256, 0, stream>>>(mask, ctrl, maskWords);
  score_kernel<<<(N_ANCH + SCORE_BLK - 1) / SCORE_BLK, SCORE_BLK, 0, stream>>>(pred, keys, aux);
  for (int r = 7; r >= 0; --r) {
    hist_kernel<<<128, 256, 0, stream>>>(keys, ctrl, r);
    select_kernel<<<1, 256, 0, stream>>>(ctrl, r);
  }
  compact_kernel<<<394, 256, 0, stream>>>(keys, ctrl, sel);
  sort_kernel<<<1, 1024, 0, stream>>>(sel);
  gather_kernel<<<(MAX_NMS + 255) / 256, 256, 0, stream>>>(sel, aux, soa);
  dim3 grid(MAX_NMS / 16 / 4, MAX_NMS / 16);   // 4 jt-subtiles per block
  iou_kernel<<<grid, 128, 0, stream>>>(soa, mask16);
  nms_out_kernel<<<1, 128, 0, stream>>>(mask, soa, out);
}